// Adaptive_Pooling_Layer_16329465660234
// MI455X (gfx1250) — compile-verified
//
#include <hip/hip_runtime.h>
#include <math.h>

// B=32, N=1024, D=64, H=8, K=128, T=12, Dout=64.  adj input is UNUSED by the reference.
typedef __attribute__((ext_vector_type(2))) float v2f;
typedef __attribute__((ext_vector_type(8))) float v8f;

#define BD   32
#define ND   1024
#define DD   64
#define HD   8
#define KD   128
#define TD   12
#define NT   8          // N / 128

// ---------------------------------------------------------------------------
// K1: x[b,n,d] = mean_t node_set[b,d,n,t]   (dominant ~100MB stream)
// node_set chunk per (b,d,n) is 12 floats = 48B, 16B-aligned -> 3x float4.
// ---------------------------------------------------------------------------
__global__ __launch_bounds__(256) void k_mean_t(const float* __restrict__ ns,
                                                float* __restrict__ x) {
    long tid = (long)blockIdx.x * 256 + threadIdx.x;          // over B*D*N, n fastest
    int n = (int)(tid & 1023);
    int d = (int)((tid >> 10) & 63);
    int b = (int)(tid >> 16);
    const float4* p = (const float4*)(ns + ((long)(b * DD + d) * ND + n) * TD);
    float4 v0 = p[0], v1 = p[1], v2 = p[2];
    float s = v0.x + v0.y + v0.z + v0.w
            + v1.x + v1.y + v1.z + v1.w
            + v2.x + v2.y + v2.z + v2.w;
    x[(long)(b * ND + n) * DD + d] = s * (1.0f / 12.0f);
}

// ---------------------------------------------------------------------------
// K2: sumx[b,d] = sum_n x ; h_avg[b,e] = tanh(sum_d (sumx/N)[d] * W0[d,e])
// ---------------------------------------------------------------------------
__global__ __launch_bounds__(64) void k_seed(const float* __restrict__ x,
                                             const float* __restrict__ W0,
                                             float* __restrict__ sumx,
                                             float* __restrict__ h_avg) {
    int b = blockIdx.x, d = threadIdx.x;
    __shared__ float temp[DD];
    const float* xb = x + (long)b * ND * DD;
    float s = 0.f;
    for (int n = 0; n < ND; ++n) s += xb[n * DD + d];
    sumx[b * DD + d] = s;
    temp[d] = s * (1.0f / (float)ND);
    __syncthreads();
    float acc = 0.f;
    for (int e = 0; e < DD; ++e) acc += temp[e] * W0[e * DD + d];
    h_avg[b * DD + d] = tanhf(acc);
}

// ---------------------------------------------------------------------------
// K3: att[n]=x[n,:].h_avg ; cent[d]=sum_n att[n]*x[n,d] ; xn[n]=||x[n,:]||
// ---------------------------------------------------------------------------
__global__ __launch_bounds__(256) void k_att_cent(const float* __restrict__ x,
                                                  const float* __restrict__ h_avg,
                                                  float* __restrict__ cent,
                                                  float* __restrict__ xn) {
    int b = blockIdx.x;
    __shared__ float att[ND];
    __shared__ float ha[DD];
    const float* xb = x + (long)b * ND * DD;
    if (threadIdx.x < DD) ha[threadIdx.x] = h_avg[b * DD + threadIdx.x];
    __syncthreads();
    for (int n = threadIdx.x; n < ND; n += 256) {
        float dot = 0.f, nrm = 0.f;
        for (int d = 0; d < DD; ++d) {
            float v = xb[n * DD + d];
            dot += v * ha[d];
            nrm += v * v;
        }
        att[n] = dot;
        xn[b * ND + n] = sqrtf(nrm);
    }
    __syncthreads();
    if (threadIdx.x < DD) {
        int d = threadIdx.x;
        float c = 0.f;
        for (int n = 0; n < ND; ++n) c += att[n] * xb[n * DD + d];
        cent[b * DD + d] = c;
    }
}

// ---------------------------------------------------------------------------
// K4: centroid MLP:
//  bc0[d,h] = relu(cent[d]*w_i2c[h] + b_i2c[h])
//  bc[b,hk,d] = relu(sum_h bc0[d,h]*W_lin[hk,h] + b_lin[hk]) ; cn[b,hk]=||.||
// ---------------------------------------------------------------------------
__global__ __launch_bounds__(256) void k_centroids(const float* __restrict__ cent,
                                                   const float* __restrict__ w_i2c,
                                                   const float* __restrict__ b_i2c,
                                                   const float* __restrict__ W_lin,
                                                   const float* __restrict__ b_lin,
                                                   float* __restrict__ bc,
                                                   float* __restrict__ cn) {
    int b = blockIdx.x;
    __shared__ float bc0[DD][HD];
    __shared__ float cs[DD];
    if (threadIdx.x < DD) cs[threadIdx.x] = cent[b * DD + threadIdx.x];
    __syncthreads();
    for (int i = threadIdx.x; i < DD * HD; i += 256) {
        int d = i >> 3, h = i & 7;
        bc0[d][h] = fmaxf(cs[d] * w_i2c[h] + b_i2c[h], 0.f);
    }
    __syncthreads();
    for (int hk = threadIdx.x; hk < HD * KD; hk += 256) {
        float wl[HD];
        #pragma unroll
        for (int h = 0; h < HD; ++h) wl[h] = W_lin[hk * HD + h];
        float bl = b_lin[hk];
        float nrm = 0.f;
        float* dst = bc + ((long)b * HD * KD + hk) * DD;
        for (int d = 0; d < DD; ++d) {
            float a = bl;
            #pragma unroll
            for (int h = 0; h < HD; ++h) a += bc0[d][h] * wl[h];
            a = fmaxf(a, 0.f);
            dst[d] = a;
            nrm += a * a;
        }
        cn[b * HD * KD + hk] = sqrtf(nrm);
    }
}

// ---------------------------------------------------------------------------
// K5: per (b, n-tile of 128): for each head h:
//   GEMM1 (WMMA f32 16x16x4): dots[128k x 128n] = bc_tile(128x64) * x_tile^T
//   cosine scale, normalize over k (full K range is inside the tile),
//   weight by conv_w[h]; GEMM2 (WMMA) accumulates pooled [128k x 64d]
//   contribution in registers across all 8 heads. One partial per (b,nt).
// A-fragments / GEMM1 B-fragments are contiguous 8B-aligned pairs in LDS ->
// load as v2f so operand pairs land directly in even-aligned VGPR pairs.
// ---------------------------------------------------------------------------
__global__ __launch_bounds__(256) void k_assign_pool(const float* __restrict__ x,
                                                     const float* __restrict__ bc,
                                                     const float* __restrict__ xn,
                                                     const float* __restrict__ cn,
                                                     const float* __restrict__ conv_w,
                                                     float* __restrict__ part) {
    __shared__ float lds_x[128 * 64];     // x tile  [n][d]
    __shared__ float lds_bc[128 * 64];    // bc tile [k][d]  (per head)
    __shared__ float lds_dots[128 * 128]; // dots    [k][n]
    __shared__ float lds_xn[128];
    __shared__ float lds_cn[128];
    __shared__ float lds_s[128];

    int b  = blockIdx.x >> 3;
    int nt = blockIdx.x & 7;
    int n0 = nt * 128;

    int w    = threadIdx.x >> 5;     // wave 0..7 -> k rows [w*16, w*16+16)
    int lane = threadIdx.x & 31;
    int l15  = lane & 15;
    int hi   = lane >> 4;            // half-wave select (K offset +2 / M offset +8)
    int mrow = (w << 4) + l15;

    // x tile + xn (persistent for all heads)
    const float* xsrc = x + ((long)b * ND + n0) * DD;
    for (int i = threadIdx.x; i < 128 * 64; i += 256) lds_x[i] = xsrc[i];
    if (threadIdx.x < 128) lds_xn[threadIdx.x] = xn[b * ND + n0 + threadIdx.x];

    v8f acc2[4] = {};                // pooled [16k x 64d] per wave, summed over heads

    for (int h = 0; h < HD; ++h) {
        __syncthreads();             // prev GEMM2 done before bc/dots overwrite
        const float* bsrc = bc + ((long)b * HD * KD + h * KD) * DD;
        for (int i = threadIdx.x; i < 128 * 64; i += 256) lds_bc[i] = bsrc[i];
        if (threadIdx.x < 128) lds_cn[threadIdx.x] = cn[b * HD * KD + h * KD + threadIdx.x];
        __syncthreads();

        // ---- GEMM1: dots[k][n] = sum_d bc[k][d] * x[n][d] ----
        v8f acc[8] = {};
        #pragma unroll
        for (int kk = 0; kk < 16; ++kk) {          // K=64 in steps of 4
            int kb = kk * 4 + hi * 2;
            v2f a = *(const v2f*)(lds_bc + mrow * 64 + kb);
            #pragma unroll
            for (int nb = 0; nb < 8; ++nb) {
                int ncol = nb * 16 + l15;
                v2f bf = *(const v2f*)(lds_x + ncol * 64 + kb);
                acc[nb] = __builtin_amdgcn_wmma_f32_16x16x4_f32(
                    false, a, false, bf, (short)0, acc[nb], false, false);
            }
        }
        #pragma unroll
        for (int nb = 0; nb < 8; ++nb)
            #pragma unroll
            for (int j = 0; j < 8; ++j)
                lds_dots[(w * 16 + hi * 8 + j) * 128 + nb * 16 + l15] = acc[nb][j];
        __syncthreads();

        // ---- cosine scale ----
        for (int i = threadIdx.x; i < 128 * 128; i += 256) {
            int k = i >> 7, n = i & 127;
            lds_dots[i] = lds_dots[i] / fmaxf(lds_cn[k] * lds_xn[n], 1e-6f);
        }
        __syncthreads();
        // ---- column sums over k (full centroid range) ----
        if (threadIdx.x < 128) {
            int n = threadIdx.x;
            float s = 0.f;
            for (int k = 0; k < 128; ++k) s += lds_dots[k * 128 + n];
            lds_s[n] = s;
        }
        __syncthreads();
        // ---- normalize + conv_w weight ----
        float cw = conv_w[h];
        for (int i = threadIdx.x; i < 128 * 128; i += 256) {
            int n = i & 127;
            lds_dots[i] = cw * lds_dots[i] / (lds_s[n] + 1e-10f);
        }
        __syncthreads();

        // ---- GEMM2: pooled[k][d] += sum_n Cw[k][n] * x[n][d]  (K=128) ----
        #pragma unroll
        for (int kk = 0; kk < 32; ++kk) {
            int kb = kk * 4 + hi * 2;
            v2f a = *(const v2f*)(lds_dots + mrow * 128 + kb);
            #pragma unroll
            for (int nb = 0; nb < 4; ++nb) {
                int dcol = nb * 16 + l15;
                v2f bf;
                bf.x = lds_x[kb * 64 + dcol];        // stride-64: compiler merges
                bf.y = lds_x[(kb + 1) * 64 + dcol];  // into ds_load_2addr_b32
                acc2[nb] = __builtin_amdgcn_wmma_f32_16x16x4_f32(
                    false, a, false, bf, (short)0, acc2[nb], false, false);
            }
        }
    }

    // write per-(b, n-tile) partial pooled block [128k x 64d]
    float* pp = part + ((long)(b * NT + nt) * KD) * DD;
    #pragma unroll
    for (int nb = 0; nb < 4; ++nb)
        #pragma unroll
        for (int j = 0; j < 8; ++j)
            pp[(w * 16 + hi * 8 + j) * 64 + nb * 16 + l15] = acc2[nb][j];
}

// ---------------------------------------------------------------------------
// K6: nn[k,d] = sum_nt part + conv_b * sumx[d] ; out = nn @ W_feat^T + b_feat
// ---------------------------------------------------------------------------
__global__ __launch_bounds__(256) void k_out(const float* __restrict__ part,
                                             const float* __restrict__ sumx,
                                             const float* __restrict__ W_feat,
                                             const float* __restrict__ b_feat,
                                             const float* __restrict__ conv_b,
                                             float* __restrict__ out) {
    int b = blockIdx.x;
    __shared__ float nn[KD * DD];
    __shared__ float sx[DD];
    if (threadIdx.x < DD) sx[threadIdx.x] = sumx[b * DD + threadIdx.x];
    __syncthreads();
    float cb = conv_b[0];
    for (int i = threadIdx.x; i < KD * DD; i += 256) {
        int d = i & 63;
        float v = cb * sx[d];
        #pragma unroll
        for (int t = 0; t < NT; ++t) v += part[((long)(b * NT + t) * KD) * DD + i];
        nn[i] = v;
    }
    __syncthreads();
    for (int i = threadIdx.x; i < KD * DD; i += 256) {
        int k = i >> 6, o = i & 63;
        float v = b_feat[o];
        for (int d = 0; d < DD; ++d) v += nn[k * DD + d] * W_feat[o * DD + d];
        out[((long)b * KD + k) * DD + o] = v;
    }
}

// ---------------------------------------------------------------------------
extern "C" void kernel_launch(void* const* d_in, const int* in_sizes, int n_in,
                              void* d_out, int out_size, void* d_ws, size_t ws_size,
                              hipStream_t stream) {
    const float* node_set = (const float*)d_in[0];
    // d_in[1] = adj : UNUSED by the reference
    const float* W0     = (const float*)d_in[2];
    const float* w_i2c  = (const float*)d_in[3];
    const float* b_i2c  = (const float*)d_in[4];
    const float* W_lin  = (const float*)d_in[5];
    const float* b_lin  = (const float*)d_in[6];
    const float* conv_w = (const float*)d_in[7];
    const float* conv_b = (const float*)d_in[8];
    const float* W_feat = (const float*)d_in[9];
    const float* b_feat = (const float*)d_in[10];
    float* out = (float*)d_out;

    // workspace carve-up (floats); total ~6.37M floats (~24.3 MB)
    float* ws    = (float*)d_ws;
    float* x     = ws;                              // B*N*D   = 2,097,152
    float* bc    = x    + (long)BD * ND * DD;       // B*HK*D  = 2,097,152
    float* part  = bc   + (long)BD * HD * KD * DD;  // B*NT*K*D= 2,097,152
    float* xn    = part + (long)BD * NT * KD * DD;  // B*N     = 32,768
    float* cn    = xn   + (long)BD * ND;            // B*HK    = 32,768
    float* sumx  = cn   + (long)BD * HD * KD;       // B*D     = 2,048
    float* h_avg = sumx + (long)BD * DD;            // B*D     = 2,048
    float* cent  = h_avg+ (long)BD * DD;            // B*D     = 2,048

    k_mean_t   <<<(BD * DD * ND) / 256, 256, 0, stream>>>(node_set, x);
    k_seed     <<<BD,  64, 0, stream>>>(x, W0, sumx, h_avg);
    k_att_cent <<<BD, 256, 0, stream>>>(x, h_avg, cent, xn);
    k_centroids<<<BD, 256, 0, stream>>>(cent, w_i2c, b_i2c, W_lin, b_lin, bc, cn);
    k_assign_pool<<<BD * NT, 256, 0, stream>>>(x, bc, xn, cn, conv_w, part);
    k_out      <<<BD, 256, 0, stream>>>(part, sumx, W_feat, b_feat, conv_b, out);
}